// SparseCodingLoss_18476949307532
// MI455X (gfx1250) — compile-verified
//
#include <hip/hip_runtime.h>
#include <hip/hip_bf16.h>

#define L_SIG   8192
#define ATOM    512
#define NATOMS  128
#define BATCH   4
#define NSTEPS  16
#define TT      8                 // position tiles (of 16) per block
#define SEG     (TT * 16 + ATOM)  // 640-float residual segment in LDS
#define TGROUPS (L_SIG / (16 * TT))  // 64 tile-groups per batch
#define KCHUNKS (ATOM / 4)        // 128 fp32-WMMA K-chunks

typedef float v2f __attribute__((ext_vector_type(2)));
typedef float v8f __attribute__((ext_vector_type(8)));

#if defined(__has_builtin)
#if __has_builtin(__builtin_amdgcn_wmma_f32_16x16x4_f32)
#define USE_WMMA_F32 1
#endif
#if __has_builtin(__builtin_amdgcn_global_load_async_to_lds_b32)
#define USE_ASYNC_LDS 1
#endif
#endif

// ---------------------------------------------------------------- helpers
__device__ __forceinline__ unsigned int ordered_of_float(float f) {
    unsigned int u = __float_as_uint(f);
    return (u & 0x80000000u) ? ~u : (u | 0x80000000u);
}
__device__ __forceinline__ float float_of_ordered(unsigned int o) {
    unsigned int u = (o & 0x80000000u) ? (o & 0x7fffffffu) : ~o;
    return __uint_as_float(u);
}

// ---------------------------------------------------------------- init
__global__ void mp_init_kernel(const float* __restrict__ recon,
                               const float* __restrict__ target,
                               float* __restrict__ res_base,
                               unsigned long long* __restrict__ packed) {
    int i = blockIdx.x * blockDim.x + threadIdx.x;
    if (i < BATCH * L_SIG) {
        res_base[i] = recon[i];
        res_base[BATCH * L_SIG + i] = target[i];
    }
    if (i < 2 * BATCH) packed[i] = 0ull;
}

// ---------------------------------------------------------------- score + argmax
// grid: (BATCH * TGROUPS, 2 signals), block: 256 (8 wave32 waves; wave w = atoms [16w,16w+16))
__global__ __launch_bounds__(256)
void mp_score_kernel(const float* __restrict__ res_base,
                     const float* __restrict__ dict,
                     unsigned long long* __restrict__ packed) {
    __shared__ float sres[SEG];
    __shared__ unsigned long long sbest;

    const int sig = blockIdx.y;
    const int b   = blockIdx.x / TGROUPS;
    const int tg  = blockIdx.x % TGROUPS;
    const int tid = threadIdx.x;
    const int tstart = tg * (16 * TT);
    const float* res = res_base + (sig * BATCH + b) * L_SIG;

    if (tid == 0) sbest = 0ull;

    // Residual segment -> LDS, zero-padded past L (matches FFT zero-pad semantics).
#ifdef USE_ASYNC_LDS
    for (int i = tid; i < SEG; i += 256) {
        int gi = tstart + i;
        if (gi < L_SIG) {
            __builtin_amdgcn_global_load_async_to_lds_b32(
                (__attribute__((address_space(1))) int*)(res + gi),
                (__attribute__((address_space(3))) int*)&sres[i],
                0, 0);
        } else {
            sres[i] = 0.0f;
        }
    }
#if __has_builtin(__builtin_amdgcn_s_wait_asynccnt)
    __builtin_amdgcn_s_wait_asynccnt(0);
#else
    asm volatile("s_wait_asynccnt 0" ::: "memory");
#endif
#else
    for (int i = tid; i < SEG; i += 256) {
        int gi = tstart + i;
        sres[i] = (gi < L_SIG) ? res[gi] : 0.0f;
    }
#endif
    __syncthreads();

    const int lane = tid & 31;          // wave32
    const int wave = tid >> 5;          // 0..7 -> atom group
    const int n    = lane & 15;
    const int h    = lane >> 4;         // K-half select per CDNA5 f32 WMMA layout
    const int atom_base = wave * 16;

    const float* drow = dict + (atom_base + n) * ATOM + 2 * h;  // A-matrix row for this lane
    const float* srow = sres + n + 2 * h;                        // Toeplitz B row base
    __builtin_prefetch(drow, 0, 3);     // global_prefetch_b8

    v8f acc[TT];
#pragma unroll
    for (int t = 0; t < TT; ++t) {
        v8f z = {0.f, 0.f, 0.f, 0.f, 0.f, 0.f, 0.f, 0.f};
        acc[t] = z;
    }

#ifdef USE_WMMA_F32
    // scores[a, t] = sum_k d[a,k] * r[t+k] as K-chunked 16x16x4 fp32 WMMA,
    // software-pipelined: fragments for chunk kc+1 load while chunk kc's 8 WMMAs issue.
    v2f afrag = {drow[0], drow[1]};     // K = 2h + {0,1}
    v2f bf[TT];
#pragma unroll
    for (int t = 0; t < TT; ++t) bf[t] = {srow[t * 16], srow[t * 16 + 1]};

#pragma unroll 2
    for (int kc = 0; kc < KCHUNKS; ++kc) {
        const int k1 = ((kc + 1) & (KCHUNKS - 1)) * 4;  // next chunk (wraps harmlessly)
        v2f anext = {drow[k1], drow[k1 + 1]};
        v2f bnext[TT];
#pragma unroll
        for (int t = 0; t < TT; ++t)
            bnext[t] = {srow[t * 16 + k1], srow[t * 16 + k1 + 1]};
#pragma unroll
        for (int t = 0; t < TT; ++t)
            acc[t] = __builtin_amdgcn_wmma_f32_16x16x4_f32(
                false, afrag, false, bf[t], (short)0, acc[t], false, false);
        afrag = anext;
#pragma unroll
        for (int t = 0; t < TT; ++t) bf[t] = bnext[t];
    }
#else
    // Compile-safety fallback (same accumulator layout), VALU only.
    for (int t = 0; t < TT; ++t) {
        for (int j = 0; j < 8; ++j) {
            const float* dr = dict + (atom_base + j + 8 * h) * ATOM;
            const float* rr = sres + t * 16 + n;
            float s = 0.f;
            for (int k = 0; k < ATOM; ++k) s += dr[k] * rr[k];
            acc[t][j] = s;
        }
    }
#endif

    // Per-lane argmax over its 8x8 slots. D layout: element j -> atom = base + j + 8h, pos = tstart + 16t + n.
    float bestv = -3.4e38f;
    unsigned int besti = 0u;
#pragma unroll
    for (int t = 0; t < TT; ++t) {
#pragma unroll
        for (int j = 0; j < 8; ++j) {
            float v = acc[t][j];
            unsigned int idx =
                (unsigned int)((atom_base + j + 8 * h) * L_SIG + (tstart + t * 16 + n));
            if (v > bestv || (v == bestv && idx < besti)) { bestv = v; besti = idx; }
        }
    }
    unsigned long long pk =
        ((unsigned long long)ordered_of_float(bestv) << 32) |
        (unsigned long long)(~besti);   // ~idx: ties pick smallest flat index (jnp.argmax)
    atomicMax(&sbest, pk);
    __syncthreads();
    if (tid == 0) atomicMax(&packed[sig * BATCH + b], sbest);
}

// ---------------------------------------------------------------- residual update + entry record
// grid: (BATCH, 2), block: 512
__global__ void mp_update_kernel(float* __restrict__ res_base,
                                 const float* __restrict__ dict,
                                 unsigned long long* __restrict__ packed,
                                 float* __restrict__ ev, int* __restrict__ ea,
                                 int* __restrict__ ep, int step) {
    const int b = blockIdx.x, sig = blockIdx.y;
    __shared__ float sv;
    __shared__ int sai, sp;
    if (threadIdx.x == 0) {
        unsigned long long pk = packed[sig * BATCH + b];
        float v = float_of_ordered((unsigned int)(pk >> 32));
        unsigned int idx = ~((unsigned int)(pk & 0xffffffffu));
        int ai = (int)(idx / L_SIG);
        int p  = (int)(idx % L_SIG);
        sv = v; sai = ai; sp = p;
        int e = (sig * BATCH + b) * NSTEPS + step;
        ev[e] = v; ea[e] = ai; ep[e] = p;
        packed[sig * BATCH + b] = 0ull;  // reset for next step
    }
    __syncthreads();
    const float v = sv;
    const int ai = sai, p = sp;
    float* res = res_base + (sig * BATCH + b) * L_SIG;
    for (int i = threadIdx.x; i < ATOM; i += blockDim.x) {
        int gi = p + i;
        if (gi < L_SIG) res[gi] -= v * dict[ai * ATOM + i];  // r_pad tail discarded
    }
}

// ---------------------------------------------------------------- sparse BCE loss
__global__ void mp_final_kernel(const float* __restrict__ ev, const int* __restrict__ ea,
                                const int* __restrict__ ep, float* __restrict__ out) {
    if (threadIdx.x != 0 || blockIdx.x != 0) return;
    long long keys[2][BATCH * NSTEPS];
    float vals[2][BATCH * NSTEPS];
    int cnt[2];
    float mx = 0.f;  // full maps are mostly zero, so array-max >= 0
    for (int s = 0; s < 2; ++s) {
        int c = 0;
        for (int e = 0; e < BATCH * NSTEPS; ++e) {
            int b = e / NSTEPS;
            int g = s * BATCH * NSTEPS + e;
            long long key = (((long long)b * 512 + ea[g]) * L_SIG) + ep[g];
            int f = -1;
            for (int q = 0; q < c; ++q)
                if (keys[s][q] == key) { f = q; break; }
            if (f >= 0) vals[s][f] += ev[g];
            else { keys[s][c] = key; vals[s][c] = ev[g]; ++c; }
        }
        cnt[s] = c;
        for (int q = 0; q < c; ++q) mx = fmaxf(mx, vals[s][q]);
    }
    double sum = 0.0;  // zero-zero cells contribute exactly 0 after the -100 clamp
    for (int q = 0; q < cnt[0]; ++q) {
        float rv = vals[0][q] / mx;
        float tv = 0.f;
        for (int w = 0; w < cnt[1]; ++w)
            if (keys[1][w] == keys[0][q]) { tv = vals[1][w] / mx; break; }
        float lr  = fmaxf(logf(rv),    -100.f);
        float l1r = fmaxf(log1pf(-rv), -100.f);
        sum += (double)(-(tv * lr + (1.f - tv) * l1r));
    }
    for (int w = 0; w < cnt[1]; ++w) {
        bool inR = false;
        for (int q = 0; q < cnt[0]; ++q)
            if (keys[0][q] == keys[1][w]) { inR = true; break; }
        if (inR) continue;
        float tv = vals[1][w] / mx;     // r==0 here: contrib = -(tv * -100) = 100*tv
        sum += (double)(100.f * tv);
    }
    out[0] = (float)(sum / (double)((long long)BATCH * 512 * L_SIG));
}

// ---------------------------------------------------------------- host launch
extern "C" void kernel_launch(void* const* d_in, const int* in_sizes, int n_in,
                              void* d_out, int out_size, void* d_ws, size_t ws_size,
                              hipStream_t stream) {
    const float* recon  = (const float*)d_in[0];
    const float* target = (const float*)d_in[1];
    const float* dict   = (const float*)d_in[2];

    char* ws = (char*)d_ws;
    unsigned long long* packed = (unsigned long long*)ws;                 // 8 * u64
    float* res = (float*)(ws + 256);                                      // 2*B*L f32
    float* ev  = (float*)(ws + 256 + (size_t)2 * BATCH * L_SIG * 4);      // 128 f32
    int*   ea  = (int*)((char*)ev + 2 * BATCH * NSTEPS * 4);              // 128 i32
    int*   ep  = (int*)((char*)ea + 2 * BATCH * NSTEPS * 4);              // 128 i32

    mp_init_kernel<<<(BATCH * L_SIG + 255) / 256, 256, 0, stream>>>(recon, target, res, packed);

    for (int step = 0; step < NSTEPS; ++step) {
        mp_score_kernel<<<dim3(BATCH * TGROUPS, 2), 256, 0, stream>>>(res, dict, packed);
        mp_update_kernel<<<dim3(BATCH, 2), 512, 0, stream>>>(res, dict, packed, ev, ea, ep, step);
    }

    mp_final_kernel<<<1, 32, 0, stream>>>(ev, ea, ep, (float*)d_out);
}